// JaxEKF_49151605735443
// MI455X (gfx1250) — compile-verified
//
#include <hip/hip_runtime.h>

// Problem constants (reference: T=1024, B=256, D=8)
#define T_STEPS 1024
#define BATCH   256
#define DSTATE  8
#define CHUNK_L 32
#define NCHUNK  32   // T_STEPS / CHUNK_L

typedef __attribute__((ext_vector_type(2))) float v2f;
typedef __attribute__((ext_vector_type(4))) float v4f;
typedef __attribute__((ext_vector_type(8))) float v8f;
typedef __attribute__((ext_vector_type(4))) unsigned int v4u;
typedef __attribute__((ext_vector_type(4))) int v4i;
typedef __attribute__((ext_vector_type(8))) int v8i;

// Workspace layout (in floats):
//  AKops: [T][32 lanes][8 slots]  -- A_t and K_t pre-swizzled as WMMA A-operands
//  Pops:  [T][32 lanes][4 slots]  -- within-chunk prefix propagators as A-operands
//  MB:    [NCHUNK][BATCH][8]      -- boundary (chunk-end) filtered means
#define OFF_AK 0
#define OFF_P  (T_STEPS * 256)
#define OFF_MB (OFF_P + T_STEPS * 128)

// Kernel-2 LDS staging (floats): x tile 32x512 = 64KB, AK slice 32x256 = 32KB
#define LDS_X_FLOATS  (CHUNK_L * 512)   // 16384
#define LDS_AK_FLOATS (CHUNK_L * 256)   // 8192
#define LDS_BYTES     ((LDS_X_FLOATS + LDS_AK_FLOATS) * 4)

static __device__ __forceinline__ v2f mk2(float a, float b) {
  v2f r; r.x = a; r.y = b; return r;
}

// D = A(16x4, f32) * B(4x16, f32) + C(16x16, f32)
static __device__ __forceinline__ v8f wmma4(v2f a, v2f b, v8f c) {
  return __builtin_amdgcn_wmma_f32_16x16x4_f32(false, a, false, b, (short)0, c,
                                               false, false);
}

static __device__ __forceinline__ float bperm_low16(int bidx, float v) {
  return __builtin_bit_cast(
      float, __builtin_amdgcn_ds_bpermute(bidx, __builtin_bit_cast(int, v)));
}

// Convert a C/D-layout accumulator (rows = 8 state dims in vgprs 0..7,
// lanes 0..15; rows 8..15 are zero) into the two B-operand register pairs:
//   bo0 = rows {0,2}/{1,3},  bo1 = rows {4,6}/{5,7}
static __device__ __forceinline__ void acc_to_b(const v8f acc, int lane, int bidx,
                                                v2f& bo0, v2f& bo1) {
  float h0 = bperm_low16(bidx, acc[2]);
  float h1 = bperm_low16(bidx, acc[3]);
  float h2 = bperm_low16(bidx, acc[6]);
  float h3 = bperm_low16(bidx, acc[7]);
  bool hi = (lane & 16) != 0;
  bo0 = mk2(hi ? h0 : acc[0], hi ? h1 : acc[1]);
  bo1 = mk2(hi ? h2 : acc[4], hi ? h3 : acc[5]);
}

// Build TDM D# groups for a 2-D fp32 tile load into LDS and issue it.
//   tile_d0 contiguous elements per row, tile_d1 rows, row stride (elements).
// This toolchain exposes the 6-arg clang-23 builtin:
//   (uint32x4 g0, int32x8 g1, int32x4, int32x4, int32x8, i32 cpol)
static __device__ __forceinline__ void tdm_load_2d(const void* gaddr,
                                                   unsigned lds_byte_off,
                                                   unsigned tile_d0,
                                                   unsigned tile_d1,
                                                   unsigned stride_d0) {
  unsigned long long ga = (unsigned long long)gaddr;
  v4u g0;
  g0[0] = 1u;                                   // count=1, user mode
  g0[1] = lds_byte_off;                         // LDS destination (bytes)
  g0[2] = (unsigned)(ga & 0xffffffffu);         // global addr [31:0]
  g0[3] = (unsigned)((ga >> 32) & 0x1ffffffu)   // global addr [56:32]
          | (2u << 30);                         // type = 2 ("image")
  v8i g1 = {0, 0, 0, 0, 0, 0, 0, 0};
  g1[0] = (int)(2u << 16);                      // data_size = 4 bytes
  g1[1] = (int)((tile_d0 & 0xffffu) << 16);     // tensor_dim0[15:0]
  g1[2] = (int)(((tile_d0 >> 16) & 0xffffu) |   // tensor_dim0[31:16]
                ((tile_d1 & 0xffffu) << 16));   // tensor_dim1[15:0]
  g1[3] = (int)(((tile_d1 >> 16) & 0xffffu) |   // tensor_dim1[31:16]
                ((tile_d0 & 0xffffu) << 16));   // tile_dim0
  g1[4] = (int)(tile_d1 & 0xffffu);             // tile_dim1 (tile_dim2 = 0)
  g1[5] = (int)stride_d0;                       // tensor_dim0_stride[31:0]
  g1[6] = 0;                                    // stride0[47:32] | stride1[15:0]
  g1[7] = 0;                                    // tensor_dim1_stride[47:16]
  v4i gz4 = {0, 0, 0, 0};                       // groups 2/3 unused (2-D tensor)
  v8i gz8 = {0, 0, 0, 0, 0, 0, 0, 0};           // extra group (unused, zero)
  __builtin_amdgcn_tensor_load_to_lds(g0, g1, gz4, gz4, gz8, 0);
}

// ---------------------------------------------------------------------------
// Kernel 1: serial Riccati recursion (batch-independent), ONE wave32 so every
// __syncthreads() is a runtime no-op (single-wave workgroup => barrier == NOP).
// Each lane owns two adjacent elements (same row) of the 8x8 matrices.
//   Pp = P + Q; S = Pp + R; K = Pp * S^-1; A = I - K; P = A * Pp
//   Pprop_t = A_t * ... * A_{chunkStart}  (reset each chunk)
// Emits A_t/K_t/Pprop_t pre-swizzled into WMMA A-operand lane layout.
// ---------------------------------------------------------------------------
__global__ void riccati_gains_kernel(const float* __restrict__ Q,
                                     const float* __restrict__ R,
                                     float* __restrict__ ws) {
  __shared__ float sQ[64], sR[64], sPp[64], sW[64], sInv[64],
                   sK[64], sA[64], sProp[64], sP[64];
  const int tid = threadIdx.x;          // 32 threads = 1 wave
  const int e0 = tid * 2, e1 = e0 + 1;  // two elements, same row
  const int i0 = e0 >> 3, j0 = e0 & 7;  // j0 even; element 1 is (i0, j0+1)

  sQ[e0] = Q[e0]; sQ[e1] = Q[e1];
  sR[e0] = R[e0]; sR[e1] = R[e1];
  sP[e0] = (i0 == j0) ? 1.0f : 0.0f;       // prior P0 = I
  sP[e1] = (i0 == j0 + 1) ? 1.0f : 0.0f;
  __syncthreads();

  for (int t = 0; t < T_STEPS; ++t) {
    float pp0 = sP[e0] + sQ[e0];
    float pp1 = sP[e1] + sQ[e1];
    sPp[e0] = pp0; sPp[e1] = pp1;
    sW[e0] = pp0 + sR[e0];
    sW[e1] = pp1 + sR[e1];
    sInv[e0] = (i0 == j0) ? 1.0f : 0.0f;
    sInv[e1] = (i0 == j0 + 1) ? 1.0f : 0.0f;
    __syncthreads();

    // Gauss-Jordan inverse of S (SPD -> no pivoting)
    for (int k = 0; k < 8; ++k) {
      float ip = 1.0f / sW[k * 8 + k];
      __syncthreads();
      if (i0 == k) {
        sW[e0] *= ip; sInv[e0] *= ip;
        sW[e1] *= ip; sInv[e1] *= ip;
      }
      __syncthreads();
      float f   = sW[i0 * 8 + k];          // shared by both elements (same row)
      float wk0 = sW[k * 8 + j0],  wk1 = sW[k * 8 + j0 + 1];
      float vk0 = sInv[k * 8 + j0], vk1 = sInv[k * 8 + j0 + 1];
      __syncthreads();
      if (i0 != k) {
        sW[e0] -= f * wk0; sInv[e0] -= f * vk0;
        sW[e1] -= f * wk1; sInv[e1] -= f * vk1;
      }
      __syncthreads();
    }

    // K = Pp * Sinv ;  A = I - K
    float kv0 = 0.0f, kv1 = 0.0f;
    for (int l = 0; l < 8; ++l) {
      float p = sPp[i0 * 8 + l];
      kv0 += p * sInv[l * 8 + j0];
      kv1 += p * sInv[l * 8 + j0 + 1];
    }
    sK[e0] = kv0; sK[e1] = kv1;
    float av0 = ((i0 == j0) ? 1.0f : 0.0f) - kv0;
    float av1 = ((i0 == j0 + 1) ? 1.0f : 0.0f) - kv1;
    sA[e0] = av0; sA[e1] = av1;
    __syncthreads();

    // P = A * Pp ; Pprop update (reset at chunk starts)
    float pn0 = 0.0f, pn1 = 0.0f;
    for (int l = 0; l < 8; ++l) {
      float a = sA[i0 * 8 + l];
      pn0 += a * sPp[l * 8 + j0];
      pn1 += a * sPp[l * 8 + j0 + 1];
    }
    float pr0, pr1;
    if ((t & (CHUNK_L - 1)) == 0) {
      pr0 = av0; pr1 = av1;
    } else {
      pr0 = 0.0f; pr1 = 0.0f;
      for (int l = 0; l < 8; ++l) {
        float a = sA[i0 * 8 + l];
        pr0 += a * sProp[l * 8 + j0];
        pr1 += a * sProp[l * 8 + j0 + 1];
      }
    }
    __syncthreads();
    sP[e0] = pn0; sP[e1] = pn1;
    sProp[e0] = pr0; sProp[e1] = pr1;
    __syncthreads();

    // A/K operands: slots 0..3 = A (2 wmma x 2 vgpr), slots 4..7 = K.
    // value(lane l, slot s) = M[l&15][ (s&3)>>1 *4 + ((l>>4)<<1) + (s&1) ]
    for (int e = tid; e < 256; e += 32) {
      int l = e >> 3, s = e & 7;
      int q = s & 3, h = q >> 1, v = q & 1;
      int row = l & 15;
      int col = h * 4 + ((l >> 4) << 1) + v;
      float val = 0.0f;
      if (row < 8) val = (s >= 4) ? sK[row * 8 + col] : sA[row * 8 + col];
      ws[OFF_AK + (size_t)t * 256 + e] = val;
    }
    // Propagator operands: 4 slots (2 wmma x 2 vgpr)
    for (int e = tid; e < 128; e += 32) {
      int l = e >> 2, s = e & 3;
      int h = s >> 1, v = s & 1;
      int row = l & 15;
      int col = h * 4 + ((l >> 4) << 1) + v;
      ws[OFF_P + (size_t)t * 128 + e] = (row < 8) ? sProp[row * 8 + col] : 0.0f;
    }
    __syncthreads();
  }
}

// ---------------------------------------------------------------------------
// Kernel 2: within-chunk local scan, batch-parallel, all math on WMMA.
// The Tensor Data Mover stages the chunk's x tile (32 x 512 fp32, row stride
// 2048) and the pre-swizzled A/K operand slice (32 x 256 fp32, contiguous)
// into LDS; per-step feeds are then ds_load_b128/b64.
// State tile kept transposed: D rows = 8 state dims, cols = 16 batches.
//   NEW = A_t * OLD + K_t * X_t   (4 chained v_wmma_f32_16x16x4_f32)
// ---------------------------------------------------------------------------
__global__ void local_scan_kernel(const float* __restrict__ x,
                                  const float* __restrict__ ws,
                                  float* __restrict__ out) {
  extern __shared__ float lds[];                // [LDS_X_FLOATS + LDS_AK_FLOATS]
  const int lane = threadIdx.x & 31;
  const int w    = threadIdx.x >> 5;            // wave in block: 0..3
  const int c    = blockIdx.y;                  // chunk 0..NCHUNK-1
  const int g0   = blockIdx.x * 64 + w * 16;    // first batch of this wave
  const int bidx = (lane & 15) << 2;

  // Stage this block's tiles via TDM (issued once, by wave 0 only).
  if (w == 0) {
    const float* xg = x + ((size_t)(c * CHUNK_L) * BATCH + blockIdx.x * 64) * 8;
    tdm_load_2d(xg, 0u, 512u, (unsigned)CHUNK_L, 2048u);
    const float* akg = ws + OFF_AK + (size_t)(c * CHUNK_L) * 256;
    tdm_load_2d(akg, (unsigned)(LDS_X_FLOATS * 4), 8192u, 1u, 8192u);
    __builtin_amdgcn_s_wait_tensorcnt(0);
  }
  __syncthreads();

  const float* xl  = lds;                  // x tile: [CHUNK_L][64 batches][8]
  const float* akl = lds + LDS_X_FLOATS;   // AK ops: [CHUNK_L][32 lanes][8]
  const int bb = w * 16 + (lane & 15);     // batch within block

  v8f acc = {0, 0, 0, 0, 0, 0, 0, 0};
  for (int lt = 0; lt < CHUNK_L; ++lt) {
    const float* ak = akl + lt * 256 + lane * 8;
    v4f a03 = *(const v4f*)ak;          // A_t operand (2 wmma)
    v4f a47 = *(const v4f*)(ak + 4);    // K_t operand (2 wmma)

    // X^T as B operand: lane -> x[lt][bb][(lane>>4)*2 (+1)]
    const float* xp = xl + lt * 512 + bb * 8 + ((lane >> 4) << 1);
    v2f bx0 = *(const v2f*)xp;          // state dims 0..3
    v2f bx1 = *(const v2f*)(xp + 4);    // state dims 4..7

    v2f bo0, bo1;
    acc_to_b(acc, lane, bidx, bo0, bo1);

    v8f d = {0, 0, 0, 0, 0, 0, 0, 0};
    d = wmma4(mk2(a03.x, a03.y), bo0, d);  // A(:,0:4) * OLD(0:4,:)
    d = wmma4(mk2(a03.z, a03.w), bo1, d);  // A(:,4:8) * OLD(4:8,:)
    d = wmma4(mk2(a47.x, a47.y), bx0, d);  // K(:,0:4) * X(0:4,:)
    d = wmma4(mk2(a47.z, a47.w), bx1, d);  // K(:,4:8) * X(4:8,:)
    acc = d;

    if (lane < 16) {  // col 'lane' = batch g0+lane; 8 state dims contiguous
      int t = c * CHUNK_L + lt;
      float* p = out + ((size_t)t * BATCH + g0 + lane) * 8;
      v4f lo; lo.x = acc[0]; lo.y = acc[1]; lo.z = acc[2]; lo.w = acc[3];
      v4f hi; hi.x = acc[4]; hi.y = acc[5]; hi.z = acc[6]; hi.w = acc[7];
      *(v4f*)p = lo;
      *(v4f*)(p + 4) = hi;
    }
  }
}

// ---------------------------------------------------------------------------
// Kernel 3: serial scan over chunk boundaries (NCHUNK steps):
//   M_c = G_c * M_{c-1} + D_c     (G_c = chunk propagator, D_c = chunk-end local)
// ---------------------------------------------------------------------------
__global__ void boundary_scan_kernel(const float* __restrict__ out,
                                     float* __restrict__ ws) {
  const int lane = threadIdx.x & 31;
  const int w    = threadIdx.x >> 5;   // 0..15 (512 threads)
  const int g0   = w * 16;
  const int bidx = (lane & 15) << 2;

  v8f acc = {0, 0, 0, 0, 0, 0, 0, 0};
  for (int c = 0; c < NCHUNK; ++c) {
    int t = c * CHUNK_L + CHUNK_L - 1;
    v4f pa = *(const v4f*)(ws + OFF_P + (size_t)t * 128 + (size_t)lane * 4);

    v8f cc = {0, 0, 0, 0, 0, 0, 0, 0};
    if (lane < 16) {
      const float* p = out + ((size_t)t * BATCH + g0 + lane) * 8;
      v4f lo = *(const v4f*)p;
      v4f hi = *(const v4f*)(p + 4);
      cc[0] = lo.x; cc[1] = lo.y; cc[2] = lo.z; cc[3] = lo.w;
      cc[4] = hi.x; cc[5] = hi.y; cc[6] = hi.z; cc[7] = hi.w;
    }

    v2f bo0, bo1;
    acc_to_b(acc, lane, bidx, bo0, bo1);

    v8f d = wmma4(mk2(pa.x, pa.y), bo0, cc);
    d     = wmma4(mk2(pa.z, pa.w), bo1, d);
    acc = d;

    if (lane < 16) {
      float* q = ws + OFF_MB + ((size_t)c * BATCH + g0 + lane) * 8;
      v4f lo; lo.x = acc[0]; lo.y = acc[1]; lo.z = acc[2]; lo.w = acc[3];
      v4f hi; hi.x = acc[4]; hi.y = acc[5]; hi.z = acc[6]; hi.w = acc[7];
      *(v4f*)q = lo;
      *(v4f*)(q + 4) = hi;
    }
  }
}

// ---------------------------------------------------------------------------
// Kernel 4: fully parallel fixup for chunks c >= 1:
//   out[t] += Pprop_t * M_{c-1}      (2 WMMAs per (t, batch-tile))
// ---------------------------------------------------------------------------
__global__ void fixup_kernel(float* __restrict__ out,
                             const float* __restrict__ ws) {
  const int lane = threadIdx.x & 31;
  const int w    = threadIdx.x >> 5;           // 0..3
  const int c    = blockIdx.z + 1;             // 1..NCHUNK-1
  const int lt   = blockIdx.y * 4 + w;         // 0..CHUNK_L-1
  const int t    = c * CHUNK_L + lt;
  const int g0   = blockIdx.x * 16;

  v4f pa = *(const v4f*)(ws + OFF_P + (size_t)t * 128 + (size_t)lane * 4);

  // M_{c-1}^T as B operand: lane -> MB[c-1][g0 + (lane&15)][(lane>>4)*2 (+1)]
  size_t mb = (size_t)OFF_MB +
              ((size_t)(c - 1) * BATCH + g0 + (lane & 15)) * 8 +
              ((lane >> 4) << 1);
  v2f b0 = *(const v2f*)(ws + mb);
  v2f b1 = *(const v2f*)(ws + mb + 4);

  v8f cc = {0, 0, 0, 0, 0, 0, 0, 0};
  if (lane < 16) {
    const float* p = out + ((size_t)t * BATCH + g0 + lane) * 8;
    v4f lo = *(const v4f*)p;
    v4f hi = *(const v4f*)(p + 4);
    cc[0] = lo.x; cc[1] = lo.y; cc[2] = lo.z; cc[3] = lo.w;
    cc[4] = hi.x; cc[5] = hi.y; cc[6] = hi.z; cc[7] = hi.w;
  }

  v8f d = wmma4(mk2(pa.x, pa.y), b0, cc);
  d     = wmma4(mk2(pa.z, pa.w), b1, d);

  if (lane < 16) {
    float* p = out + ((size_t)t * BATCH + g0 + lane) * 8;
    v4f lo; lo.x = d[0]; lo.y = d[1]; lo.z = d[2]; lo.w = d[3];
    v4f hi; hi.x = d[4]; hi.y = d[5]; hi.z = d[6]; hi.w = d[7];
    *(v4f*)p = lo;
    *(v4f*)(p + 4) = hi;
  }
}

extern "C" void kernel_launch(void* const* d_in, const int* in_sizes, int n_in,
                              void* d_out, int out_size, void* d_ws,
                              size_t ws_size, hipStream_t stream) {
  (void)in_sizes; (void)n_in; (void)out_size; (void)ws_size;
  const float* x = (const float*)d_in[0];   // [T, B, D] fp32
  const float* Q = (const float*)d_in[1];   // [D, D] fp32
  const float* R = (const float*)d_in[2];   // [D, D] fp32
  float* out = (float*)d_out;               // [T, B, D] fp32
  float* ws  = (float*)d_ws;

  // 1) batch-independent gains + propagators (serial over T, single wave)
  riccati_gains_kernel<<<dim3(1), dim3(32), 0, stream>>>(Q, R, ws);
  // 2) within-chunk local scans: 32 chunks x 4 batch-groups, TDM-staged LDS
  local_scan_kernel<<<dim3(4, NCHUNK), dim3(128), LDS_BYTES, stream>>>(x, ws, out);
  // 3) serial scan over the 32 chunk boundaries (16 waves)
  boundary_scan_kernel<<<dim3(1), dim3(512), 0, stream>>>(out, ws);
  // 4) parallel fixup of chunks 1..NCHUNK-1
  fixup_kernel<<<dim3(16, CHUNK_L / 4, NCHUNK - 1), dim3(128), 0, stream>>>(out, ws);
}